// MyDropout_20899310863270
// MI455X (gfx1250) — compile-verified
//
#include <hip/hip_runtime.h>
#include <stdint.h>

// -----------------------------------------------------------------------------
// MyDropout "attack branch": out = x * mask
//   mask = dropout(ones with cols<64 zeroed, p=0.1), JAX threefry2x32 key (0,42)
// Memory-bound (512 MB traffic, ~22 us floor @ 23.3 TB/s) but with ~37 int-VALU
// ops/element of PRNG work -> use gfx1250 async global->LDS copies to overlap
// the byte stream with threefry compute. Wave32, no barriers (per-thread LDS
// slots, per-wave ASYNCcnt ordering).
// -----------------------------------------------------------------------------

#define THREADS 256
#define VEC 4      // float4 per half per thread per iteration
#define ITER 4     // pipelined iterations per block

typedef float v4f __attribute__((ext_vector_type(4)));  // native clang vector

// ---- s_wait_asynccnt: builtin if the toolchain has it, else inline asm ------
#if defined(__has_builtin)
#  if __has_builtin(__builtin_amdgcn_s_wait_asynccnt)
#    define WAIT_ASYNCCNT(n) __builtin_amdgcn_s_wait_asynccnt(n)
#  endif
#endif
#ifndef WAIT_ASYNCCNT
#  define WAIT_ASYNCCNT(n) asm volatile("s_wait_asynccnt %0" ::"i"(n) : "memory")
#endif

// ---- async global -> LDS 128-bit copy (CDNA5 GLOBAL_LOAD_ASYNC_TO_LDS_B128) -
// Flat pointers to LDS carry the wave-relative LDS byte offset in addr[31:0]
// (ISA 10.2 aperture rules), so the low 32 bits are exactly the VDST LDS addr.
__device__ __forceinline__ void async_copy_b128(const float* __restrict__ gptr,
                                                float* lptr) {
  uint32_t lds_addr = (uint32_t)(uintptr_t)lptr;
  asm volatile("global_load_async_to_lds_b128 %0, %1, off"
               :
               : "v"(lds_addr), "v"(gptr)
               : "memory");
}

// ---- JAX threefry2x32, key = (0, 42): ks2 = 0 ^ 42 ^ 0x1BD11BDA -------------
__device__ __forceinline__ uint32_t rotl32(uint32_t x, int r) {
  return (x << r) | (x >> (32 - r));
}

__device__ __forceinline__ void threefry2x32_k042(uint32_t c0, uint32_t c1,
                                                  uint32_t& o0, uint32_t& o1) {
  const uint32_t ks1 = 42u;
  const uint32_t ks2 = 0x1BD11BF0u;  // 0 ^ 42 ^ 0x1BD11BDA
  uint32_t x0 = c0;                  // + ks0 (== 0)
  uint32_t x1 = c1 + ks1;
#define TF_R(r)            \
  {                        \
    x0 += x1;              \
    x1 = rotl32(x1, (r));  \
    x1 ^= x0;              \
  }
  TF_R(13) TF_R(15) TF_R(26) TF_R(6)
  x0 += ks1; x1 += ks2 + 1u;
  TF_R(17) TF_R(29) TF_R(16) TF_R(24)
  x0 += ks2; x1 += 0u + 2u;          // ks0 == 0
  TF_R(13) TF_R(15) TF_R(26) TF_R(6)
  x0 += 0u;  x1 += ks1 + 3u;         // ks0 == 0
  TF_R(17) TF_R(29) TF_R(16) TF_R(24)
  x0 += ks1; x1 += ks2 + 4u;
  TF_R(13) TF_R(15) TF_R(26) TF_R(6)
  x0 += ks2; x1 += 0u + 5u;          // ks0 == 0
#undef TF_R
  o0 = x0;
  o1 = x1;
}

// bits -> uniform [0,1) exactly as jax.random.uniform(f32); keep if u < 0.9
__device__ __forceinline__ float mask_from_bits(uint32_t bits) {
  const float scale = (float)(1.0 / (1.0 - 0.1));  // 1/0.9 rounded like jnp
  float u = __uint_as_float((bits >> 9) | 0x3f800000u) - 1.0f;
  return (u < 0.9f) ? scale : 0.0f;
}

__global__ __launch_bounds__(THREADS) void MyDropout_attack_kernel(
    const float* __restrict__ x, float* __restrict__ out, unsigned n) {
  const unsigned half = n >> 1;  // JAX pairs element i with i + n/2
  // [buffer 0/1][low/high half][THREADS*VEC floats] = 16 KB
  __shared__ __align__(16) float lds[2][2][THREADS * VEC];

  const unsigned tid = threadIdx.x;
  const unsigned span = THREADS * VEC;  // low-half elements per iter per block
  const unsigned base0 = (blockIdx.x * ITER) * span + tid * VEC;

  // ---- pipeline prologue: stage iteration 0 ----
  if (base0 < half) {
    async_copy_b128(x + base0, &lds[0][0][tid * VEC]);
    async_copy_b128(x + base0 + half, &lds[0][1][tid * VEC]);
  }

#pragma unroll
  for (int t = 0; t < ITER; ++t) {
    const unsigned i = base0 + (unsigned)t * span;
    // ---- stage iteration t+1 (2 async ops), then drain down to them ----
    if (t + 1 < ITER) {
      const unsigned inext = i + span;
      if (inext < half) {
        const int b = (t + 1) & 1;
        async_copy_b128(x + inext, &lds[b][0][tid * VEC]);
        async_copy_b128(x + inext + half, &lds[b][1][tid * VEC]);
        WAIT_ASYNCCNT(2);  // iteration t's copies complete; t+1 in flight
      } else {
        WAIT_ASYNCCNT(0);
      }
    } else {
      WAIT_ASYNCCNT(0);
    }

    if (i >= half) continue;

    const int b = t & 1;
    const v4f xl = *(const v4f*)&lds[b][0][tid * VEC];
    const v4f xh = *(const v4f*)&lds[b][1][tid * VEC];

    // col < 64 forces zero; chunk is 4-aligned and 2^25 % 4096 == 0, so the
    // condition is uniform across the chunk and shared by the i+half partner.
    const bool colzero = (i & 4095u) < 64u;

    uint32_t bl[VEC], bh[VEC];
#pragma unroll
    for (int j = 0; j < VEC; ++j) {
      threefry2x32_k042(i + (unsigned)j, i + (unsigned)j + half, bl[j], bh[j]);
    }

    v4f ol, oh;
    float ml0 = colzero ? 0.0f : mask_from_bits(bl[0]);
    float ml1 = colzero ? 0.0f : mask_from_bits(bl[1]);
    float ml2 = colzero ? 0.0f : mask_from_bits(bl[2]);
    float ml3 = colzero ? 0.0f : mask_from_bits(bl[3]);
    float mh0 = colzero ? 0.0f : mask_from_bits(bh[0]);
    float mh1 = colzero ? 0.0f : mask_from_bits(bh[1]);
    float mh2 = colzero ? 0.0f : mask_from_bits(bh[2]);
    float mh3 = colzero ? 0.0f : mask_from_bits(bh[3]);
    ol.x = xl.x * ml0; ol.y = xl.y * ml1; ol.z = xl.z * ml2; ol.w = xl.w * ml3;
    oh.x = xh.x * mh0; oh.y = xh.y * mh1; oh.z = xh.z * mh2; oh.w = xh.w * mh3;

    // streaming 256 MB output: non-temporal stores, keep L2 for the read side
    __builtin_nontemporal_store(ol, (v4f*)(out + i));
    __builtin_nontemporal_store(oh, (v4f*)(out + i + half));
  }
}

extern "C" void kernel_launch(void* const* d_in, const int* in_sizes, int n_in,
                              void* d_out, int out_size, void* d_ws,
                              size_t ws_size, hipStream_t stream) {
  (void)n_in; (void)d_ws; (void)ws_size; (void)out_size;
  const float* x = (const float*)d_in[0];
  float* out = (float*)d_out;
  const unsigned n = (unsigned)in_sizes[0];     // 16384*4096 = 2^26
  const unsigned half = n >> 1;
  const unsigned perBlock = THREADS * VEC * ITER;  // low-half elems per block
  const unsigned blocks = (half + perBlock - 1) / perBlock;  // 8192
  MyDropout_attack_kernel<<<dim3(blocks), dim3(THREADS), 0, stream>>>(x, out, n);
}